// InterfaceBoundaryLoss_29815662969154
// MI455X (gfx1250) — compile-verified
//
#include <hip/hip_runtime.h>
#include <hip/hip_bf16.h>

typedef __attribute__((ext_vector_type(2))) float v2f;
typedef __attribute__((ext_vector_type(8))) float v8f;

#define NG 128
#define FOUR_PI_F 12.566370614359172f
#define E_IN_F  1.0f
#define E_OUT_F 80.0f
#define EPS_F   1.1920928955078125e-07f   // float32 machine eps (2^-23)
#define DX_F    0.007874015748031496f     // 1.0/127.0

// ---------------------------------------------------------------------------
// Kernel 1: pairwise Green's-function sums via V_WMMA_F32_16X16X4_F32.
// One wave (32 threads) per tile of 16 boundary points; loops over 4 tiles of
// 16 charges. Two WMMAs per (k-tile, j-tile): r^2 (with C = |p|^2 as the
// accumulator so D = r^2 directly) and x_k . n_j for the dipole term.
// ---------------------------------------------------------------------------
__global__ __launch_bounds__(32)
void green_wmma_kernel(const float* __restrict__ q,       // [64]
                       const float* __restrict__ xq,      // [64,3]
                       const float* __restrict__ points,  // [N,3]
                       const float* __restrict__ nx,
                       const float* __restrict__ ny,
                       const float* __restrict__ nz,
                       float* __restrict__ Gb,            // [N] out
                       float* __restrict__ gGn,           // [N] out
                       int N)
{
    const int lane = threadIdx.x;    // 0..31
    const int m    = lane & 15;      // row-within-half
    const int hi   = lane >> 4;      // 0: K=0,1 / M=v ; 1: K=2,3 / M=v+8
    const int j    = blockIdx.x * 16 + m;
    const int jc   = (j < N) ? j : (N - 1);   // clamp (keeps EXEC all-ones)

    // Point-side (N dimension) data
    const float px = points[3*jc + 0];
    const float py = points[3*jc + 1];
    const float pz = points[3*jc + 2];
    const float nxv = nx[jc], nyv = ny[jc], nzv = nz[jc];
    const float pp    = px*px + py*py + pz*pz;
    const float pdotn = px*nxv + py*nyv + pz*nzv;

    // B (4x16, K x N): lanes 0-15 hold K=0..1 in VGPR0..1, lanes 16-31 K=2..3
    v2f bR; bR.x = hi ? pz   : px;   bR.y = hi ? 1.0f : py;   // [px,py,pz,1]
    v2f bN; bN.x = hi ? nzv  : nxv;  bN.y = hi ? 0.0f : nyv;  // [nx,ny,nz,0]

    // C = |p_j|^2 broadcast: every C/D element (v,lane) has N = lane&15
    v8f cR;
#pragma unroll
    for (int v = 0; v < 8; ++v) cR[v] = pp;
    v8f cZ = {};

    float acc1 = 0.0f;   // sum_k q_k / r
    float acc2 = 0.0f;   // sum_k (-q_k / r^3) * (p - x_k).n

#pragma unroll
    for (int kt = 0; kt < 4; ++kt) {
        // A (16x4, M x K): lane holds row M = m, K = {0,1} or {2,3} by half
        const int c = kt * 16 + m;
        const float xx = xq[3*c + 0], xy = xq[3*c + 1], xz = xq[3*c + 2];
        const float xsq = xx*xx + xy*xy + xz*xz;

        v2f aR;  // charge row for r^2: [-2x, -2y, -2z, |x|^2]
        aR.x = hi ? (-2.0f * xz) : (-2.0f * xx);
        aR.y = hi ? xsq          : (-2.0f * xy);

        v2f aN;  // charge row for x.n: [x, y, z, 0]
        aN.x = hi ? xz   : xx;
        aN.y = hi ? 0.0f : xy;

        // D = A*B + C  ->  r^2[k][j]  and  (x_k . n_j)
        v8f r2  = __builtin_amdgcn_wmma_f32_16x16x4_f32(
                      false, aR, false, bR, (short)0, cR, false, false);
        v8f xdn = __builtin_amdgcn_wmma_f32_16x16x4_f32(
                      false, aN, false, bN, (short)0, cZ, false, false);

        // D layout: element (v, lane) is (M = v + 8*hi, N = lane&15)
#pragma unroll
        for (int v = 0; v < 8; ++v) {
            const int   ci  = kt * 16 + v + 8 * hi;  // charge index
            const float qv  = q[ci];
            const float r2v = r2[v];
            const float s   = sqrtf(r2v);
            const float r   = fmaxf(s, EPS_F);             // max(sqrt(r2), eps)
            const float r3  = fmaxf(r2v * s, EPS_F);       // max(r2^1.5, eps)
            acc1 += qv / r;
            acc2 += (-qv / r3) * (pdotn - xdn[v]);
        }
    }

    // Each lane summed M = {0..7}+8*hi for its N; combine the two halves.
    acc1 += __shfl_xor(acc1, 16, 32);
    acc2 += __shfl_xor(acc2, 16, 32);

    if (hi == 0 && j < N) {
        const float scl = 1.0f / (FOUR_PI_F * E_IN_F);
        Gb[j]  = acc1 * scl;
        gGn[j] = acc2 * scl;
    }
}

// ---------------------------------------------------------------------------
// Kernel 2: gather finite differences + squared errors, reduce into acc[2].
// One thread per (batch, point); wave-level shuffle reduction then atomicAdd.
// ---------------------------------------------------------------------------
__global__ __launch_bounds__(256)
void loss_gather_kernel(const float* __restrict__ output,  // [8,2,128,128,128]
                        const float* __restrict__ nx,
                        const float* __restrict__ ny,
                        const float* __restrict__ nz,
                        const int*   __restrict__ x_idx,
                        const int*   __restrict__ y_idx,
                        const int*   __restrict__ z_idx,
                        const float* __restrict__ Gb,
                        const float* __restrict__ gGn,
                        float* __restrict__ acc,           // [2]
                        int N)
{
    const int t     = blockIdx.x * blockDim.x + threadIdx.x;
    const int total = 8 * N;
    float e1sq = 0.0f, e2sq = 0.0f;

    if (t < total) {
        const int b = t / N;
        const int j = t - b * N;

        const int xi = x_idx[j], yi = y_idx[j], zi = z_idx[j];
        const long long cbase =
            ((((long long)(b * 2) * NG + xi) * NG + yi) * NG) + zi;
        const float* ui = output + cbase;                       // u_in center
        const float* uo = output + cbase + (long long)NG*NG*NG; // u_out center

        const float b_in  = ui[0];
        const float b_out = uo[0];
        const float nxv = nx[j], nyv = ny[j], nzv = nz[j];

        const int SX = NG * NG, SY = NG, SZ = 1;

        const float gx_in  = (nxv > 0.0f) ? (b_in - ui[-SX]) / DX_F
                                          : (ui[ SX] - b_in) / DX_F;
        const float gx_out = (nxv > 0.0f) ? (uo[ SX] - b_out) / DX_F
                                          : (b_out - uo[-SX]) / DX_F;
        const float gy_in  = (nyv > 0.0f) ? (b_in - ui[-SY]) / DX_F
                                          : (ui[ SY] - b_in) / DX_F;
        const float gy_out = (nyv > 0.0f) ? (uo[ SY] - b_out) / DX_F
                                          : (b_out - uo[-SY]) / DX_F;
        const float gz_in  = (nzv > 0.0f) ? (b_in - ui[-SZ]) / DX_F
                                          : (ui[ SZ] - b_in) / DX_F;
        const float gz_out = (nzv > 0.0f) ? (uo[ SZ] - b_out) / DX_F
                                          : (b_out - uo[-SZ]) / DX_F;

        const float dn_in  = gx_in  * nxv + gy_in  * nyv + gz_in  * nzv;
        const float dn_out = gx_out * nxv + gy_out * nyv + gz_out * nzv;

        const float e1 = b_in + Gb[j] - b_out;
        const float e2 = E_IN_F * (dn_in + gGn[j]) - E_OUT_F * dn_out;
        e1sq = e1 * e1;
        e2sq = e2 * e2;
    }

    // wave32 shuffle reduction
#pragma unroll
    for (int off = 16; off > 0; off >>= 1) {
        e1sq += __shfl_xor(e1sq, off, 32);
        e2sq += __shfl_xor(e2sq, off, 32);
    }
    if ((threadIdx.x & 31) == 0) {
        atomicAdd(&acc[0], e1sq);
        atomicAdd(&acc[1], e2sq);
    }
}

__global__ void init_acc_kernel(float* acc) {
    if (threadIdx.x < 2) acc[threadIdx.x] = 0.0f;
}

__global__ void finalize_kernel(const float* __restrict__ acc,
                                float* __restrict__ out, int N) {
    if (threadIdx.x == 0) {
        const float inv = 1.0f / (float)(8 * N);
        out[0] = (acc[0] + acc[1]) * inv;   // (loss1 + loss2) * WEIGHT(=1)
    }
}

// ---------------------------------------------------------------------------
extern "C" void kernel_launch(void* const* d_in, const int* in_sizes, int n_in,
                              void* d_out, int out_size, void* d_ws, size_t ws_size,
                              hipStream_t stream) {
    const float* output = (const float*)d_in[0];
    const float* q      = (const float*)d_in[1];
    const float* xq     = (const float*)d_in[2];
    const float* points = (const float*)d_in[3];
    const float* nx     = (const float*)d_in[4];
    const float* ny     = (const float*)d_in[5];
    const float* nz     = (const float*)d_in[6];
    const int*   x_idx  = (const int*)d_in[7];
    const int*   y_idx  = (const int*)d_in[8];
    const int*   z_idx  = (const int*)d_in[9];

    const int N = in_sizes[4];   // number of boundary points

    float* Gb  = (float*)d_ws;
    float* gGn = Gb + N;
    float* acc = gGn + N;

    init_acc_kernel<<<1, 32, 0, stream>>>(acc);

    const int jt = (N + 15) / 16;
    green_wmma_kernel<<<jt, 32, 0, stream>>>(q, xq, points, nx, ny, nz,
                                             Gb, gGn, N);

    const int total  = 8 * N;
    const int blocks = (total + 255) / 256;
    loss_gather_kernel<<<blocks, 256, 0, stream>>>(output, nx, ny, nz,
                                                   x_idx, y_idx, z_idx,
                                                   Gb, gGn, acc, N);

    finalize_kernel<<<1, 32, 0, stream>>>(acc, (float*)d_out, N);
}